// PathologySpecificLoss_42202348650721
// MI455X (gfx1250) — compile-verified
//
#include <hip/hip_runtime.h>
#include <hip/hip_bf16.h>
#include <stdint.h>

typedef float v2f __attribute__((ext_vector_type(2)));
typedef float v8f __attribute__((ext_vector_type(8)));

#define TILE_ROWS 1024
#define BLOCK 256
#define CHUNKS_PER_THREAD 5 /* 1024 rows * 20B / 16B / 256 threads */

// Low 32 bits of a generic pointer to __shared__ == workgroup-relative LDS byte
// offset (addrspacecast(3->0) is {aperture_hi, lds_offset} on AMDGPU).
__device__ __forceinline__ uint32_t lds_byte_offset(const void* p) {
  return (uint32_t)(uintptr_t)p;
}

// Issue one tile (1024 rows * 20B = 1280 x b128 chunks) of predictions into an
// LDS buffer via the CDNA5 async-to-LDS path (tracked by ASYNCcnt).
__device__ __forceinline__ void issue_tile(const char* predBase, uint32_t tileByte,
                                           uint32_t limit, float* lbuf, uint32_t tid) {
#pragma unroll
  for (int k = 0; k < CHUNKS_PER_THREAD; ++k) {
    uint32_t c = (uint32_t)k * BLOCK + tid;
    uint32_t goff = tileByte + c * 16u;
    if (goff > limit) goff = limit;           // clamp OOB tail chunks (rows guarded later)
    uint32_t laddr = lds_byte_offset(lbuf) + c * 16u;
    asm volatile("global_load_async_to_lds_b128 %0, %1, %2"
                 :
                 : "v"(laddr), "v"(goff), "s"(predBase)
                 : "memory");
  }
}

__global__ __launch_bounds__(BLOCK) void ce_partial_kernel(
    const float* __restrict__ pred, const unsigned long long* __restrict__ tgt,
    const float* __restrict__ wtab, float* __restrict__ partials, uint32_t nRows) {
  __shared__ float pbuf[2][TILE_ROWS * 5];  // 2 x 20 KB double buffer
  __shared__ float red[BLOCK];
  __shared__ float wlds[8];

  const uint32_t tid = threadIdx.x;
  const uint32_t G = gridDim.x;
  if (tid < 5u) wlds[tid] = wtab[tid];

  const uint32_t nTiles = (nRows + TILE_ROWS - 1) / TILE_ROWS;
  const uint32_t t0 = blockIdx.x;
  const uint32_t myTiles = (t0 < nTiles) ? ((nTiles - t0 + G - 1) / G) : 0u;

  const char* predBase = (const char*)pred;
  const uint32_t predBytes = nRows * 20u;
  const uint32_t predLimit = (predBytes >= 16u) ? ((predBytes - 16u) & ~15u) : 0u;
  const uint32_t tgtBytes = nRows * 8u;
  const uint32_t tgtLimit = (tgtBytes >= 32u) ? ((tgtBytes - 32u) & ~31u) : 0u;

  float acc = 0.0f;

  if (myTiles > 0u) {
    uint32_t tile = t0;
    issue_tile(predBase, tile * (TILE_ROWS * 20u), predLimit, pbuf[0], tid);
    int cur = 0;
    for (uint32_t i = 0; i < myTiles; ++i) {
      const bool hasNext = (i + 1u < myTiles);
      if (hasNext) {
        issue_tile(predBase, (tile + G) * (TILE_ROWS * 20u), predLimit, pbuf[cur ^ 1], tid);
        asm volatile("s_wait_asynccnt 0x5" ::: "memory");  // current tile's 5 loads done
      } else {
        asm volatile("s_wait_asynccnt 0x0" ::: "memory");
      }
      __syncthreads();  // all waves' async loads for this tile are visible

      // Targets: 4 consecutive int64 per thread, two b128 loads, low words only.
      uint32_t rowBase = tile * TILE_ROWS + tid * 4u;
      uint32_t tbyte = rowBase * 8u;
      if (tbyte > tgtLimit) tbyte = tgtLimit;
      const uint4* tp = (const uint4*)((const char*)tgt + tbyte);
      uint4 ta = tp[0];
      uint4 tb = tp[1];
      uint32_t tv[4] = {ta.x, ta.z, tb.x, tb.z};

      const float* rp = &pbuf[cur][tid * 20u];
#pragma unroll
      for (int j = 0; j < 4; ++j) {
        uint32_t row = rowBase + (uint32_t)j;
        if (row < nRows) {
          float p0 = rp[j * 5 + 0], p1 = rp[j * 5 + 1], p2 = rp[j * 5 + 2];
          float p3 = rp[j * 5 + 3], p4 = rp[j * 5 + 4];
          float m = fmaxf(fmaxf(fmaxf(p0, p1), fmaxf(p2, p3)), p4);
          float s = __expf(p0 - m) + __expf(p1 - m) + __expf(p2 - m) +
                    __expf(p3 - m) + __expf(p4 - m);
          float lse = m + __logf(s);
          uint32_t t = tv[j];
          if (t > 4u) t = 4u;
          float pt = (t == 0u) ? p0 : (t == 1u) ? p1 : (t == 2u) ? p2 : (t == 3u) ? p3 : p4;
          acc += wlds[t] * (lse - pt);
        }
      }
      __syncthreads();  // everyone done reading pbuf[cur] before it is refilled
      tile += G;
      cur ^= 1;
    }
  }

  // Deterministic block tree reduction.
  red[tid] = acc;
  __syncthreads();
#pragma unroll
  for (int s = BLOCK / 2; s > 0; s >>= 1) {
    if (tid < (uint32_t)s) red[tid] += red[tid + s];
    __syncthreads();
  }
  if (tid == 0u) partials[blockIdx.x] = red[0];
}

// Finalize: 256 threads gather the <=2048 block partials with 4 independent
// accumulator chains (loads can all be in flight at once, single wait), LDS
// tree down to 32 lanes, then wave 0 collapses 32->1 with one
// V_WMMA_F32_16X16X4_F32 (B = ones -> D rows are A row sums) + one shuffle.
__global__ __launch_bounds__(BLOCK) void ce_finalize_kernel(
    const float* __restrict__ partials, float* __restrict__ out,
    uint32_t G, float invN) {
  __shared__ float red[BLOCK];
  const uint32_t tid = threadIdx.x;

  float a0 = 0.0f, a1 = 0.0f, a2 = 0.0f, a3 = 0.0f;
  uint32_t i = tid;
  for (; i + 3u * BLOCK < G; i += 4u * BLOCK) {
    a0 += partials[i];
    a1 += partials[i + BLOCK];
    a2 += partials[i + 2u * BLOCK];
    a3 += partials[i + 3u * BLOCK];
  }
  for (; i < G; i += BLOCK) a0 += partials[i];

  red[tid] = (a0 + a1) + (a2 + a3);
  __syncthreads();
#pragma unroll
  for (int s = BLOCK / 2; s >= 32; s >>= 1) {
    if (tid < (uint32_t)s) red[tid] += red[tid + s];
    __syncthreads();
  }

  if (tid < 32u) {  // wave-uniform: only wave 0 executes, EXEC all ones inside
    float a = red[tid];
    v2f A;
    A.x = a;      // lane<16: A[lane][0]; lane>=16: A[lane-16][2]
    A.y = 0.0f;   // K=1 / K=3 zero
    v2f B;
    B.x = 1.0f;   // B = all ones 4x16
    B.y = 1.0f;
    v8f C = {0.f, 0.f, 0.f, 0.f, 0.f, 0.f, 0.f, 0.f};
    v8f D = __builtin_amdgcn_wmma_f32_16x16x4_f32(
        /*neg_a=*/false, A, /*neg_b=*/false, B, /*c_mod=*/(short)0, C,
        /*reuse_a=*/false, /*reuse_b=*/false);

    float s = D[0] + D[1] + D[2] + D[3] + D[4] + D[5] + D[6] + D[7];
    // lanes 0-15 hold partials{0-7,16-23}; lanes 16-31 hold partials{8-15,24-31}
    float tot = s + __shfl_xor(s, 16, 32);
    if (tid == 0u) out[0] = tot * invN;
  }
}

extern "C" void kernel_launch(void* const* d_in, const int* in_sizes, int n_in,
                              void* d_out, int out_size, void* d_ws, size_t ws_size,
                              hipStream_t stream) {
  const float* pred = (const float*)d_in[0];
  const unsigned long long* tgt = (const unsigned long long*)d_in[1];
  const float* wtab = (const float*)d_in[2];
  float* out = (float*)d_out;

  uint32_t n = (uint32_t)in_sizes[1];  // number of rows / targets
  uint32_t nTiles = (n + TILE_ROWS - 1) / TILE_ROWS;
  uint32_t G = (nTiles < 2048u) ? nTiles : 2048u;
  if (G == 0u) G = 1u;

  float* partials = (float*)d_ws;  // G floats (<= 8 KB)

  ce_partial_kernel<<<dim3(G), dim3(BLOCK), 0, stream>>>(pred, tgt, wtab, partials, n);
  float invN = (float)(1.0 / (double)n);
  ce_finalize_kernel<<<dim3(1), dim3(BLOCK), 0, stream>>>(partials, out, G, invN);
}